// GIN_46531675685231
// MI455X (gfx1250) — compile-verified
//
#include <hip/hip_runtime.h>
#include <hip/hip_bf16.h>

// GIN (2-layer GINConv, eps=0) for MI455X / gfx1250.
//   t = x + scatter_add(x[src] -> dst)
//   h = relu(t @ W1 + b1)
//   t = h + scatter_add(h[src] -> dst)
//   out = t @ W2 + b2
// Scatter-add is the bandwidth-dominant stage (~1.3 GB atomic traffic);
// the 128x128 GEMMs run on V_WMMA_F32_16X16X4_F32 (fp32-faithful WMMA).

typedef float v2f __attribute__((ext_vector_type(2)));
typedef float v8f __attribute__((ext_vector_type(8)));

#define D 128            // feature dim (in = hid = out = 128)
#define LDS_PITCH 132    // 128 + 4 pad -> conflict-free strided LDS reads

// ---------------------------------------------------------------- copy (t = src)
__global__ void gin_copy_f4(const float4* __restrict__ in,
                            float4* __restrict__ out, int n4) {
    int i = blockIdx.x * blockDim.x + threadIdx.x;
    if (i < n4) out[i] = in[i];
}

// ------------------------------------------------- scatter-add: agg[dst] += feat[src]
// One thread per (edge, feature): lanes within a 128-thread span cover one row,
// so both the row read and the float atomics are fully coalesced.
__global__ void gin_scatter_add(const float* __restrict__ feat,
                                const int* __restrict__ edges,  // [2, nEdges]
                                float* __restrict__ agg, int nEdges) {
    long long t = (long long)blockIdx.x * blockDim.x + threadIdx.x;
    if (t >= (long long)nEdges * D) return;
    int e = (int)(t >> 7);          // edge id
    int f = (int)(t & (D - 1));     // feature id
    int s = edges[e];               // src row
    int d = edges[nEdges + e];      // dst row
    atomicAdd(&agg[(long long)d * D + f], feat[(long long)s * D + f]);
}

// ------------------------------------------------------------- GEMM via fp32 WMMA
// One block per 16-row M tile. The block stages the 16x128 A tile in LDS once;
// each of the 8 waves owns one 16-column N tile and chains 32 WMMA_F32_16X16X4_F32.
//
// Fragment layouts (CDNA5 ISA 7.12.2, wave32):
//   A 16x4 f32 : lane<16 -> M=lane,    VGPR0/1 = K=k,k+1
//                lane>=16 -> M=lane-16, VGPR0/1 = K=k+2,k+3
//   B 4x16 f32 : N = lane%16, K = vgpr + 2*(lane>=16)   (mirror of A)
//   C/D 16x16  : VGPR r -> row = r + 8*(lane>=16), col = lane%16
__global__ void __launch_bounds__(256)
gin_gemm_wmma(const float* __restrict__ T,   // [nRows, 128] activations
              const float* __restrict__ W,   // [128, 128] row-major (K, N)
              const float* __restrict__ bias,// [128]
              float* __restrict__ O,         // [nRows, 128]
              int doRelu) {
    __shared__ float tileT[16 * LDS_PITCH];

    const int mBase = blockIdx.x * 16;

    // Cooperative load of the 16x128 A tile (256 threads x 8 floats).
#pragma unroll
    for (int i = 0; i < 8; ++i) {
        int idx = threadIdx.x + i * 256;
        int r = idx >> 7;        // 0..15
        int c = idx & 127;       // 0..127
        tileT[r * LDS_PITCH + c] = T[(size_t)(mBase + r) * D + c];
    }
    __syncthreads();

    const int wave = threadIdx.x >> 5;   // 0..7 -> N tile
    const int lane = threadIdx.x & 31;
    const int half = lane >> 4;          // 0 | 1
    const int l16  = lane & 15;
    const int nBase = wave * 16;
    const int n = nBase + l16;

    v8f acc = {};
#pragma unroll 4
    for (int k = 0; k < D; k += 4) {
        const int ka = k + 2 * half;
        // A fragment from LDS: (l16*132 + ka) is even -> 8B-aligned float2 load.
        v2f a = *reinterpret_cast<const v2f*>(&tileT[l16 * LDS_PITCH + ka]);
        // B fragment from global (W hot in WGP$/L2: only 64 KB).
        v2f b;
        b.x = W[(size_t)ka * D + n];
        b.y = W[(size_t)(ka + 1) * D + n];
        acc = __builtin_amdgcn_wmma_f32_16x16x4_f32(
            /*neg_a=*/false, a, /*neg_b=*/false, b,
            /*c_mod=*/(short)0, acc, /*reuse_a=*/false, /*reuse_b=*/false);
    }

    const float bv = bias[n];
#pragma unroll
    for (int r = 0; r < 8; ++r) {
        int row = mBase + r + 8 * half;
        float v = acc[r] + bv;
        if (doRelu) v = fmaxf(v, 0.0f);
        O[(size_t)row * D + n] = v;
    }
}

// --------------------------------------------------------------------- launcher
extern "C" void kernel_launch(void* const* d_in, const int* in_sizes, int n_in,
                              void* d_out, int out_size, void* d_ws, size_t ws_size,
                              hipStream_t stream) {
    const float* x     = (const float*)d_in[0];
    const int*   edges = (const int*)d_in[1];    // [2, nEdges]
    const float* W1    = (const float*)d_in[2];
    const float* b1    = (const float*)d_in[3];
    const float* W2    = (const float*)d_in[4];
    const float* b2    = (const float*)d_in[5];

    const int nNodes = in_sizes[0] / D;          // 100000
    const int nEdges = in_sizes[1] / 2;          // 640000

    float* t = (float*)d_ws;                         // [nNodes, 128]
    float* h = t + (size_t)nNodes * D;               // [nNodes, 128]

    const int n4 = (nNodes * D) / 4;
    const int cpGrid = (n4 + 255) / 256;
    const long long sTot = (long long)nEdges * D;
    const int scGrid = (int)((sTot + 255) / 256);
    const int mTiles = nNodes / 16;                  // 6250 (exact)

    // Layer 1
    gin_copy_f4<<<cpGrid, 256, 0, stream>>>((const float4*)x, (float4*)t, n4);
    gin_scatter_add<<<scGrid, 256, 0, stream>>>(x, edges, t, nEdges);
    gin_gemm_wmma<<<mTiles, 256, 0, stream>>>(t, W1, b1, h, /*relu=*/1);

    // Layer 2
    gin_copy_f4<<<cpGrid, 256, 0, stream>>>((const float4*)h, (float4*)t, n4);
    gin_scatter_add<<<scGrid, 256, 0, stream>>>(h, edges, t, nEdges);
    gin_gemm_wmma<<<mTiles, 256, 0, stream>>>(t, W2, b2, (float*)d_out, /*relu=*/0);
}